// AL2Loss2d_19516331393756
// MI455X (gfx1250) — compile-verified
//
#include <hip/hip_runtime.h>
#include <hip/hip_bf16.h>
#include <math.h>

#define NUM_CLASSES 19
#define C_DIM 64
#define B_DIM 8
#define HW (512 * 512)
#define CHUNKS 16
#define CHUNK_LEN (HW / CHUNKS)          // 16384 pixels per block
#define EPS 1e-8f

typedef __attribute__((ext_vector_type(2))) float v2f;
typedef __attribute__((ext_vector_type(8))) float v8f;

// ---------------------------------------------------------------------------
// Kernel 0: zero the accumulator workspace (19*64 sums + 19 counts floats).
// ---------------------------------------------------------------------------
__global__ void zero_ws_kernel(float* __restrict__ ws, int n) {
    int i = blockIdx.x * blockDim.x + threadIdx.x;
    if (i < n) ws[i] = 0.0f;
}

// ---------------------------------------------------------------------------
// Kernel 1: segmented scatter-sum.
// grid = (CHUNKS, C_DIM, B_DIM), block = 256.
// Each block: one (batch, channel) plane chunk. Coalesced float4 loads of the
// input plane + int4 loads of labels; ds_add_f32 into 19-entry LDS bins;
// 19 global f32 atomics at the end. Blocks with c==0 also accumulate counts.
// ---------------------------------------------------------------------------
__global__ void __launch_bounds__(256)
seg_sum_kernel(const float* __restrict__ in, const int* __restrict__ tgt,
               float* __restrict__ ws_sums /*[19][64]*/,
               float* __restrict__ ws_counts /*[19]*/) {
    __shared__ float s_acc[NUM_CLASSES];
    __shared__ float s_cnt[NUM_CLASSES];

    const int tid = threadIdx.x;
    const int c = blockIdx.y;
    const int b = blockIdx.z;
    const bool do_count = (c == 0);

    if (tid < NUM_CLASSES) {
        s_acc[tid] = 0.0f;
        s_cnt[tid] = 0.0f;
    }
    __syncthreads();

    const size_t plane = (size_t)(b * C_DIM + c) * HW + (size_t)blockIdx.x * CHUNK_LEN;
    const size_t tbase = (size_t)b * HW + (size_t)blockIdx.x * CHUNK_LEN;
    const float4* __restrict__ s4 = (const float4*)(in + plane);
    const int4* __restrict__ t4 = (const int4*)(tgt + tbase);

    const int n4 = CHUNK_LEN / 4;   // 4096 float4 per block
    for (int i = tid; i < n4; i += 256) {
        float4 v = s4[i];
        int4 k = t4[i];
        atomicAdd(&s_acc[k.x], v.x);
        atomicAdd(&s_acc[k.y], v.y);
        atomicAdd(&s_acc[k.z], v.z);
        atomicAdd(&s_acc[k.w], v.w);
        if (do_count) {
            atomicAdd(&s_cnt[k.x], 1.0f);
            atomicAdd(&s_cnt[k.y], 1.0f);
            atomicAdd(&s_cnt[k.z], 1.0f);
            atomicAdd(&s_cnt[k.w], 1.0f);
        }
    }
    __syncthreads();

    if (tid < NUM_CLASSES) {
        atomicAdd(&ws_sums[tid * C_DIM + c], s_acc[tid]);
        if (do_count) atomicAdd(&ws_counts[tid], s_cnt[tid]);
    }
}

// ---------------------------------------------------------------------------
// Kernel 2: centroids -> normalize -> 32x32 Gram via V_WMMA_F32_16X16X4_F32
// -> CosineEmbeddingLoss reduction. One block, 4 waves (128 threads).
// Wave w computes Gram tile (ti, tj) = (w>>1, w&1).
// ---------------------------------------------------------------------------
__global__ void __launch_bounds__(128)
finalize_kernel(const float* __restrict__ ws_sums,
                const float* __restrict__ ws_counts,
                float* __restrict__ out) {
    __shared__ float cl[32 * C_DIM];   // zero-padded normalized centroids [32][64]
    __shared__ float rn[32];           // 1/norm per row
    __shared__ float loss_acc;

    const int tid = threadIdx.x;

    // centers[k][c] = sums[k][c] / max(counts[k], 1); rows 19..31 = 0
    for (int idx = tid; idx < 32 * C_DIM; idx += 128) {
        const int k = idx >> 6;
        float v = 0.0f;
        if (k < NUM_CLASSES) {
            v = ws_sums[idx] / fmaxf(ws_counts[k], 1.0f);
        }
        cl[idx] = v;
    }
    if (tid == 0) loss_acc = 0.0f;
    __syncthreads();

    // 1 / max(||center_k||, eps) per row
    if (tid < 32) {
        float ss = 0.0f;
        #pragma unroll
        for (int c = 0; c < C_DIM; ++c) {
            float x = cl[tid * C_DIM + c];
            ss += x * x;
        }
        rn[tid] = 1.0f / fmaxf(sqrtf(ss), EPS);
    }
    __syncthreads();

    for (int idx = tid; idx < 32 * C_DIM; idx += 128) {
        cl[idx] *= rn[idx >> 6];
    }
    __syncthreads();

    // Gram tile via WMMA: A = cn (16x4 slabs), B = cn^T (4x16 slabs).
    // A layout (ISA 7.12.2): lanes 0-15 hold {K=k0, K=k0+1} for M=lane,
    // lanes 16-31 hold {K=k0+2, K=k0+3} for M=lane-16. B mirrors with N=lane.
    const int wave = tid >> 5;
    const int lane = tid & 31;
    const int ti = wave >> 1;
    const int tj = wave & 1;
    const int half = lane >> 4;        // 0: lanes 0-15, 1: lanes 16-31
    const int l = lane & 15;
    const int rowA = ti * 16 + l;
    const int rowB = tj * 16 + l;
    const int koff = half * 2;

    v8f acc = {};
    #pragma unroll
    for (int k0 = 0; k0 < C_DIM; k0 += 4) {
        v2f a, b;
        a.x = cl[rowA * C_DIM + k0 + koff];
        a.y = cl[rowA * C_DIM + k0 + koff + 1];
        b.x = cl[rowB * C_DIM + k0 + koff];
        b.y = cl[rowB * C_DIM + k0 + koff + 1];
        acc = __builtin_amdgcn_wmma_f32_16x16x4_f32(
            /*neg_a=*/false, a, /*neg_b=*/false, b,
            /*c_mod=*/(short)0, acc, /*reuse_a=*/false, /*reuse_b=*/false);
    }

    // C layout: VGPR r -> M = r (lanes 0-15) or r+8 (lanes 16-31), N = lane&15.
    float local = 0.0f;
    #pragma unroll
    for (int r = 0; r < 8; ++r) {
        const int row = ti * 16 + r + half * 8;
        const int col = tj * 16 + l;
        if (row < NUM_CLASSES && col < NUM_CLASSES) {
            const float s = acc[r];
            local += (row == col) ? (1.0f - s) : fmaxf(s, 0.0f);
        }
    }
    atomicAdd(&loss_acc, local);
    __syncthreads();

    if (tid == 0) {
        out[0] = loss_acc *
                 (1.0f / (float)(NUM_CLASSES * NUM_CLASSES * NUM_CLASSES));
    }
}

// ---------------------------------------------------------------------------
extern "C" void kernel_launch(void* const* d_in, const int* in_sizes, int n_in,
                              void* d_out, int out_size, void* d_ws, size_t ws_size,
                              hipStream_t stream) {
    (void)in_sizes; (void)n_in; (void)out_size; (void)ws_size;

    const float* in = (const float*)d_in[0];
    const int* tgt = (const int*)d_in[1];
    float* out = (float*)d_out;

    float* ws = (float*)d_ws;
    float* ws_sums = ws;                               // [19][64]
    float* ws_counts = ws + NUM_CLASSES * C_DIM;       // [19]
    const int n_ws = NUM_CLASSES * C_DIM + NUM_CLASSES;

    zero_ws_kernel<<<(n_ws + 255) / 256, 256, 0, stream>>>(ws, n_ws);

    seg_sum_kernel<<<dim3(CHUNKS, C_DIM, B_DIM), 256, 0, stream>>>(
        in, tgt, ws_sums, ws_counts);

    finalize_kernel<<<1, 128, 0, stream>>>(ws_sums, ws_counts, out);
}